// DifferentiableHistogram_29549374996798
// MI455X (gfx1250) — compile-verified
//
#include <hip/hip_runtime.h>
#include <math.h>

// DifferentiableHistogram for MI455X (gfx1250, wave32).
//
// Phase 1: per-workgroup LDS scatter histogram (ds_add_f32 atomics), partials to d_ws.
// Phase 2: reduce G partials per batch as a ones-GEMM via V_WMMA_F32_16X16X4_F32
//          (exact fp32), then normalize sqrt(N / (sum(N)+1e-8)).

#define NBINS   64
#define NB2     (NBINS * NBINS)      // 4096
#define HLO     (-2.85f)
#define HHI     ( 2.85f)
#define EPSV    (1e-8f)

typedef __attribute__((ext_vector_type(2))) float v2f;
typedef __attribute__((ext_vector_type(8))) float v8f;

// ---------------------------------------------------------------------------
// Kernel 1: scatter histogram. grid = (G, B), block = 256 (8 wave32 waves).
// Each workgroup strides the whole batch-plane and produces one 64x64 partial.
// ---------------------------------------------------------------------------
__global__ __launch_bounds__(256)
void hist_partial_kernel(const float* __restrict__ img,
                         float* __restrict__ ws,
                         int P, int G)
{
    __shared__ float hist[NB2];            // 16 KB of 320 KB/WGP LDS

    const int tid = threadIdx.x;
    const int wg  = blockIdx.x;            // 0..G-1
    const int bat = blockIdx.y;            // 0..B-1

    for (int i = tid; i < NB2; i += blockDim.x) hist[i] = 0.0f;
    __syncthreads();

    const float eps_bin = (HHI - HLO) / (float)(NBINS - 1);
    const float inv_eps = 1.0f / eps_bin;
    const float* base = img + (size_t)bat * 3 * (size_t)P;

    for (int p = wg * 256 + tid; p < P; p += G * 256) {
        // coalesced channel-plane reads (stride P between channels)
        const float r  = base[p];
        const float g  = base[(size_t)P + p];
        const float bl = base[2 * (size_t)P + p];

        const float s = r + g + bl;
        if (!(s > EPSV)) continue;                      // valid mask (Iy would be 0)
        const float Iy = sqrtf(r * r + g * g + bl * bl);

        const float lr = logf(r  + EPSV);
        const float lg = logf(g  + EPSV);
        const float lb = logf(bl + EPSV);
        const float u  = lg - lr;
        const float v  = lg - lb;

        // u-bins: A_u[i] = HLO + eps_bin*i ; mask: 0 < |u-A_u[i]| <= eps_bin
        // candidates are integers in [tu-1, tu+1] => subset of {i0-1, i0, i0+1}
        int   iu[4]; int nu = 0;
        {
            const float tu = (u - HLO) * inv_eps;
            const int   i0 = (int)floorf(tu);
            #pragma unroll
            for (int d = -1; d <= 2; ++d) {
                const int i = i0 + d;
                if ((unsigned)i < (unsigned)NBINS) {
                    const float du = fabsf(u - (HLO + eps_bin * (float)i));
                    if (du > 0.0f && du <= eps_bin) iu[nu++] = i;
                }
            }
        }
        if (nu == 0) continue;

        // v-bins: A_v[j] = HLO + eps_bin*(63-j)  (reversed centers)
        int   jv[4]; int nv = 0;
        {
            const float tj = 63.0f - (v - HLO) * inv_eps;
            const int   j0 = (int)floorf(tj);
            #pragma unroll
            for (int d = -1; d <= 2; ++d) {
                const int j = j0 + d;
                if ((unsigned)j < (unsigned)NBINS) {
                    const float dv = fabsf(v - (HLO + eps_bin * (float)(NBINS - 1 - j)));
                    if (dv > 0.0f && dv <= eps_bin) jv[nv++] = j;
                }
            }
        }

        // <=4 LDS float atomics per pixel (ds_add_f32)
        for (int jj = 0; jj < nv; ++jj) {
            const int rowoff = jv[jj] * NBINS;
            for (int ii = 0; ii < nu; ++ii) {
                atomicAdd(&hist[rowoff + iu[ii]], Iy);
            }
        }
    }
    __syncthreads();

    // stream full partial (zeros included -> no ws pre-init needed)
    float* dst = ws + ((size_t)bat * G + wg) * NB2;
    for (int i = tid; i < NB2; i += blockDim.x) dst[i] = hist[i];
}

// ---------------------------------------------------------------------------
// Kernel 2: reduce G partials per batch with V_WMMA_F32_16X16X4_F32 as a
// ones-GEMM colsum, then normalize. grid = (B), block = 256 (8 waves).
//
// WMMA f32 16x16x4 layouts (wave32):
//   A (16x4): lane = row M (mod 16), VGPR0/1 = K{0,1} (lanes 0-15) / K{2,3}.
//             A = all-ones, layout-invariant.
//   B (4x16): lane = col N (mod 16), VGPR0/1 = K{0,1} (lanes 0-15) / K{2,3}.
//   D (16x16): VGPR0 lanes hold row M=0 (and M=8 in upper half); all 16 output
//             rows are identical column sums, so c[0] is valid in every lane.
// ---------------------------------------------------------------------------
__global__ __launch_bounds__(256)
void reduce_norm_kernel(const float* __restrict__ ws,
                        float* __restrict__ out,
                        int G)
{
    __shared__ float Nsh[NB2];             // 16 KB
    __shared__ float red[8];

    const int tid  = threadIdx.x;
    const int bat  = blockIdx.x;
    const int wave = tid >> 5;
    const int lane = tid & 31;

    const float* pb = ws + (size_t)bat * G * NB2;

    v2f a; a.x = 1.0f; a.y = 1.0f;         // ones A-matrix

    // 256 column tiles of 16, distributed over 8 waves
    for (int tile = wave; tile < NB2 / 16; tile += 8) {
        const int col = tile * 16 + (lane & 15);
        const int kh  = (lane < 16) ? 0 : 2;   // K-half held by this lane
        v8f c = {};
        for (int k = 0; k < G; k += 4) {
            v2f bb;
            bb.x = pb[(size_t)(k + kh)     * NB2 + col];
            bb.y = pb[(size_t)(k + kh + 1) * NB2 + col];
            // D = ones(16x4) * B(4x16) + C  -> every D row = colsum of 4 partials
            c = __builtin_amdgcn_wmma_f32_16x16x4_f32(
                    /*neg_a=*/false, a, /*neg_b=*/false, bb,
                    /*c_mod=*/(short)0, c, /*reuse_a=*/false, /*reuse_b=*/false);
        }
        if (lane < 16) Nsh[col] = c[0] + 1e-8f;   // N = einsum + 1e-8
    }
    __syncthreads();

    // norm = sum(N) + 1e-8
    float partial = 0.0f;
    for (int i = tid; i < NB2; i += blockDim.x) partial += Nsh[i];
    #pragma unroll
    for (int off = 16; off > 0; off >>= 1)
        partial += __shfl_down(partial, off, 32);
    if (lane == 0) red[wave] = partial;
    __syncthreads();
    if (tid == 0) {
        float t = 0.0f;
        #pragma unroll
        for (int w = 0; w < 8; ++w) t += red[w];
        red[0] = t;
    }
    __syncthreads();

    const float inv = 1.0f / (red[0] + 1e-8f);
    float* dst = out + (size_t)bat * NB2;
    for (int i = tid; i < NB2; i += blockDim.x)
        dst[i] = sqrtf(Nsh[i] * inv);
}

// ---------------------------------------------------------------------------
extern "C" void kernel_launch(void* const* d_in, const int* in_sizes, int n_in,
                              void* d_out, int out_size, void* d_ws, size_t ws_size,
                              hipStream_t stream)
{
    const float* img = (const float*)d_in[0];
    float*       out = (float*)d_out;
    float*       ws  = (float*)d_ws;

    const int B = out_size / NB2;                 // 4
    const int P = in_sizes[0] / (3 * B);          // 384*512 = 196608

    // partials per batch, sized to workspace; multiple of 4 for the WMMA K-loop
    const size_t per_partial = (size_t)NB2 * sizeof(float);
    long long gmax = (long long)(ws_size / (per_partial * (size_t)B));
    int G = 64;
    if ((long long)G > gmax) G = (int)gmax;
    G &= ~3;
    if (G < 4) G = 4;

    hist_partial_kernel<<<dim3(G, B), 256, 0, stream>>>(img, ws, P, G);
    reduce_norm_kernel<<<dim3(B), 256, 0, stream>>>(ws, out, G);
}